// VectorQuantizer_42271068127610
// MI455X (gfx1250) — compile-verified
//
#include <hip/hip_runtime.h>

typedef __attribute__((ext_vector_type(2))) float v2f;
typedef __attribute__((ext_vector_type(8))) float v8f;

#define VQ_N       65536      // 64*32*32 rows
#define VQ_D       64
#define VQ_K       1024
#define VQ_BETA    0.25f
#define ROWS_PB    64                       // rows per block (4 M-tiles)
#define NBLOCKS    (VQ_N / ROWS_PB)         // 1024
#define LDS_STRIDE 68                       // padded row stride (floats)

// ---------------------------------------------------------------------------
// Kernel 0: codebook squared norms |c_k|^2  -> ws
// ---------------------------------------------------------------------------
__global__ void vq_csq_kernel(const float* __restrict__ cb, float* __restrict__ csq) {
    int k = blockIdx.x * blockDim.x + threadIdx.x;
    if (k < VQ_K) {
        const float4* row = (const float4*)(cb + (size_t)k * VQ_D);
        float s = 0.f;
#pragma unroll
        for (int i = 0; i < VQ_D / 4; ++i) {
            float4 v = row[i];
            s += v.x * v.x + v.y * v.y + v.z * v.z + v.w * v.w;
        }
        csq[k] = s;
    }
}

// ---------------------------------------------------------------------------
// Kernel 1: 64-row tile per block. Column-tile outer loop keeps the 16 B
// fragments in registers and reuses them across 4 M-tiles. A tile staged in
// LDS (padded stride 68 -> conflict-free ds_load_b64 fragment reads).
// fp32 WMMA 16x16x4 accumulation; fully branchless argmin updates.
// Block = 256 threads = 8 waves; wave w owns column tiles ct*8 + w.
// ---------------------------------------------------------------------------
__global__ __launch_bounds__(256) void vq_main_kernel(
        const float* __restrict__ ze,       // [N, 64]
        const float* __restrict__ cb,       // [K, 64]
        const float* __restrict__ csq,      // [K]
        float* __restrict__ out_idx,        // [N] (indices as float)
        float* __restrict__ out_zq,         // [N, 64]
        float* __restrict__ partial) {      // [NBLOCKS]
    __shared__ float s_A[ROWS_PB * LDS_STRIDE];   // 64x68 floats, 17 KB
    __shared__ float s_minv[8][ROWS_PB];
    __shared__ int   s_mini[8][ROWS_PB];
    __shared__ int   s_idx[ROWS_PB];
    __shared__ float s_red[256];

    const int tid   = threadIdx.x;
    const int lane  = tid & 31;
    const int wave  = tid >> 5;           // 0..7
    const int m0    = blockIdx.x * ROWS_PB;
    const int lhalf = lane >> 4;          // 0: K pair {4k,4k+1}; 1: {4k+2,4k+3}
    const int l16   = lane & 15;

    // --- stage A tile (64 rows x 64 floats) into padded LDS ---
    {
        const int r = tid >> 2;           // 0..63
        const int q = (tid & 3) * 16;     // 0,16,32,48
        const float4* src = (const float4*)(ze + (size_t)(m0 + r) * VQ_D + q);
        float* dst = s_A + r * LDS_STRIDE + q;
#pragma unroll
        for (int i = 0; i < 4; ++i)
            *(float4*)(dst + 4 * i) = src[i];
    }
    __syncthreads();

    // running argmin of (|c|^2 - 2*S) per (M-tile, accumulator row)
    float minv[4][8];
    int   mini[4][8];
#pragma unroll
    for (int mt = 0; mt < 4; ++mt)
#pragma unroll
        for (int v = 0; v < 8; ++v) { minv[mt][v] = 3.4e38f; mini[mt][v] = 0x7fffffff; }

    for (int ct = 0; ct < 8; ++ct) {
        const int col = (ct * 8 + wave) * 16 + l16;
        // B fragment: B[k][n] = cb[col][k]; lane = N, VGPR+half = K
        const float* brow = cb + (size_t)col * VQ_D + 2 * lhalf;
        v2f bfrag[16];
#pragma unroll
        for (int k = 0; k < 16; ++k)
            bfrag[k] = *(const v2f*)(brow + 4 * k);
        const float cs = csq[col];

#pragma unroll
        for (int mt = 0; mt < 4; ++mt) {
            // A fragment from LDS: lane = M (mod 16), VGPR+half = K
            const float* alds = s_A + (mt * 16 + l16) * LDS_STRIDE + 2 * lhalf;
            v8f c = {};
#pragma unroll
            for (int k = 0; k < 16; ++k) {
                v2f a = *(const v2f*)(alds + 4 * k);
                c = __builtin_amdgcn_wmma_f32_16x16x4_f32(
                        /*neg_a=*/false, a, /*neg_b=*/false, bfrag[k],
                        /*c_mod=*/(short)0, c, /*reuse_a=*/false, /*reuse_b=*/false);
            }
            // Branchless min update. Within a lane, col strictly increases
            // with ct, so strict '<' alone preserves first-index tie-break.
#pragma unroll
            for (int v = 0; v < 8; ++v) {
                float d = __builtin_fmaf(-2.0f, c[v], cs);
                bool p = d < minv[mt][v];
                minv[mt][v] = p ? d   : minv[mt][v];
                mini[mt][v] = p ? col : mini[mt][v];
            }
        }
    }

    // --- reduce across the 16 lanes sharing rows (stays within wave halves);
    //     branchless with index tie-break ---
#pragma unroll
    for (int off = 8; off >= 1; off >>= 1)
#pragma unroll
        for (int mt = 0; mt < 4; ++mt)
#pragma unroll
            for (int v = 0; v < 8; ++v) {
                float mv = minv[mt][v];
                int   mi = mini[mt][v];
                float ov = __shfl_xor(mv, off, 32);
                int   oi = __shfl_xor(mi, off, 32);
                bool p = (ov < mv) | ((ov == mv) & (oi < mi));
                minv[mt][v] = p ? ov : mv;
                mini[mt][v] = p ? oi : mi;
            }

    if (l16 == 0) {          // lane 0 (rows v) and lane 16 (rows v+8) per M-tile
#pragma unroll
        for (int mt = 0; mt < 4; ++mt)
#pragma unroll
            for (int v = 0; v < 8; ++v) {
                s_minv[wave][mt * 16 + v + 8 * lhalf] = minv[mt][v];
                s_mini[wave][mt * 16 + v + 8 * lhalf] = mini[mt][v];
            }
    }
    __syncthreads();

    // --- reduce across the 8 waves; first-index tie-break like jnp.argmin ---
    if (tid < ROWS_PB) {
        float bv = s_minv[0][tid];
        int   bi = s_mini[0][tid];
#pragma unroll
        for (int w = 1; w < 8; ++w) {
            float v = s_minv[w][tid];
            int   i = s_mini[w][tid];
            bool p = (v < bv) | ((v == bv) & (i < bi));
            bv = p ? v : bv;
            bi = p ? i : bi;
        }
        s_idx[tid] = bi;
        out_idx[m0 + tid] = (float)bi;
    }
    __syncthreads();

    // --- gather codebook rows, write z_q, accumulate (e - z)^2 (z from LDS) ---
    {
        const int r   = tid >> 2;         // 0..63
        const int q   = (tid & 3) * 16;   // 0,16,32,48
        const int idx = s_idx[r];
        const float4* erow = (const float4*)(cb + (size_t)idx * VQ_D + q);
        const float*  zrow = s_A + r * LDS_STRIDE + q;
        float4* orow = (float4*)(out_zq + (size_t)(m0 + r) * VQ_D + q);
        float ss = 0.f;
#pragma unroll
        for (int i = 0; i < 4; ++i) {
            float4 e = erow[i];
            float4 z = *(const float4*)(zrow + 4 * i);
            orow[i] = e;
            float dx = e.x - z.x, dy = e.y - z.y, dz = e.z - z.z, dw = e.w - z.w;
            ss += dx * dx + dy * dy + dz * dz + dw * dw;
        }
        s_red[tid] = ss;
    }
    __syncthreads();
#pragma unroll
    for (int s = 128; s >= 1; s >>= 1) {
        if (tid < s) s_red[tid] += s_red[tid + s];
        __syncthreads();
    }
    if (tid == 0) partial[blockIdx.x] = s_red[0];
}

// ---------------------------------------------------------------------------
// Kernel 2: reduce per-block partials -> vq_loss scalar
// vq_loss = (1 + BETA) * sum / (N * D)
// ---------------------------------------------------------------------------
__global__ void vq_loss_kernel(const float* __restrict__ partial,
                               float* __restrict__ loss_out) {
    __shared__ float s[256];
    float acc = 0.f;
    for (int i = threadIdx.x; i < NBLOCKS; i += 256) acc += partial[i];
    s[threadIdx.x] = acc;
    __syncthreads();
#pragma unroll
    for (int st = 128; st >= 1; st >>= 1) {
        if (threadIdx.x < st) s[threadIdx.x] += s[threadIdx.x + st];
        __syncthreads();
    }
    if (threadIdx.x == 0)
        *loss_out = (1.0f + VQ_BETA) * s[0] / (float)(VQ_N * VQ_D);
}

// ---------------------------------------------------------------------------
extern "C" void kernel_launch(void* const* d_in, const int* in_sizes, int n_in,
                              void* d_out, int out_size, void* d_ws, size_t ws_size,
                              hipStream_t stream) {
    const float* ze = (const float*)d_in[0];   // [65536, 64]
    const float* cb = (const float*)d_in[1];   // [1024, 64]

    float* out      = (float*)d_out;
    float* out_idx  = out;                       // 65536 indices (as float)
    float* out_zq   = out + VQ_N;                // 65536*64
    float* out_loss = out + VQ_N + VQ_N * VQ_D;  // scalar

    float* csq     = (float*)d_ws;               // 1024 floats
    float* partial = csq + VQ_K;                 // NBLOCKS floats

    vq_csq_kernel<<<(VQ_K + 255) / 256, 256, 0, stream>>>(cb, csq);
    vq_main_kernel<<<NBLOCKS, 256, 0, stream>>>(ze, cb, csq, out_idx, out_zq, partial);
    vq_loss_kernel<<<1, 256, 0, stream>>>(partial, out_loss);
}